// AdaptiveMemoryBank_74414603371128
// MI455X (gfx1250) — compile-verified
//
#include <hip/hip_runtime.h>
#include <hip/hip_bf16.h>

// Problem constants (match reference)
#define B_ROWS 4096
#define M_ROWS 65536
#define DIM    1024
#define TOPK   32
#define NSPLIT 4
#define STRIPE (M_ROWS / NSPLIT)   // 16384 columns per workgroup stripe
#define TM     64                  // query rows per workgroup (sim kernel)
#define NT     16                  // bank columns per tile
#define KCHUNK 256                 // K staging chunk in LDS

typedef __attribute__((ext_vector_type(2))) float v2f;
typedef __attribute__((ext_vector_type(8))) float v8f;

#define NEG_INF (-3.402823466e38f)

// ---------------------------------------------------------------------------
// CDNA5 async global->LDS helpers (ASYNCcnt-tracked, no VGPR round trip)
// ---------------------------------------------------------------------------
__device__ __forceinline__ void async_b128(const float* gsrc, float* ldst)
{
    unsigned lds = (unsigned)(size_t)ldst;     // addrspace(3) offset
    asm volatile("global_load_async_to_lds_b128 %0, %1, off"
                 :: "v"(lds), "v"(gsrc) : "memory");
}
__device__ __forceinline__ void wait_async0()
{
    asm volatile("s_wait_asynccnt 0" ::: "memory");
}

// Copy a bank tile [NT rows x KCHUNK floats] (global row stride DIM) into LDS.
// 1024 float4 slots across 128 threads -> 8 async b128 issues per thread.
__device__ __forceinline__ void issue_bank_chunk(const float* __restrict__ gbase,
                                                 float* __restrict__ sdst, int tid)
{
#pragma unroll
    for (int j = 0; j < 8; ++j) {
        const int i   = tid + j * 128;   // float4 slot index
        const int row = i >> 6;          // KCHUNK/4 = 64 float4 per row
        const int off = i & 63;
        async_b128(gbase + (size_t)row * DIM + (size_t)off * 4, sdst + (size_t)i * 4);
    }
}

// ---------------------------------------------------------------------------
// Generic f32 WMMA GEMM: C[rows x DIM] = A[rows x DIM] @ op(W) (+ bias)
//   TRANSB=1 : C = A @ W^T   (W row-major [N, K])   -> used for q @ Wk^T + bk
//   TRANSB=0 : C = A @ W     (W row-major [K, N])   -> used for qk @ Wk
// One wave computes one 16x16 output tile over full K=1024.
// grid = (rows/16, DIM/64), block = 128 (4 waves)
// ---------------------------------------------------------------------------
template <bool TRANSB, bool BIAS>
__global__ __launch_bounds__(128)
void proj_gemm_kernel(const float* __restrict__ A, const float* __restrict__ W,
                      const float* __restrict__ bias, float* __restrict__ C)
{
    const int wave  = threadIdx.x >> 5;
    const int lane  = threadIdx.x & 31;
    const int m0    = blockIdx.x * 16;
    const int n0    = (blockIdx.y * 4 + wave) * 16;
    const int l15   = lane & 15;
    const int khalf = (lane >> 4) * 2;      // A/B frag: VGPR j holds K = khalf + j

    const size_t arow = (size_t)(m0 + l15) * DIM;
    const size_t brow = (size_t)(n0 + l15) * DIM;

    v8f acc = {};
    for (int k0 = 0; k0 < DIM; k0 += 4) {
        const int kk = k0 + khalf;
        v2f a = *(const v2f*)(A + arow + kk);
        v2f b;
        if (TRANSB) {
            b = *(const v2f*)(W + brow + kk);
        } else {
            b[0] = W[(size_t)kk * DIM + n0 + l15];
            b[1] = W[(size_t)(kk + 1) * DIM + n0 + l15];
        }
        acc = __builtin_amdgcn_wmma_f32_16x16x4_f32(
            false, a, false, b, (short)0, acc, false, false);
    }

    // C/D layout: lane holds column n = l15, rows m = 8*(lane>>4)+j
    const int   mbase = m0 + (lane >> 4) * 8;
    const int   n     = n0 + l15;
    const float bv    = BIAS ? bias[n] : 0.0f;
#pragma unroll
    for (int j = 0; j < 8; ++j)
        C[(size_t)(mbase + j) * DIM + n] = acc[j] + bv;
}

// ---------------------------------------------------------------------------
// Fused similarity GEMM + streaming top-32.
// grid = (B_ROWS/TM, NSPLIT), block = 128 (4 waves; wave w owns rows 16w..16w+15)
// LDS: A block (TM x DIM f32, resident, async-loaded) + double-buffered bank
//      tile staging (async) + per-wave tile spill + per-row top-k lists.
//      256 + 32 + 4 + 16 = 308 KB dynamic LDS (1 WG / WGP).
// ---------------------------------------------------------------------------
__global__ __launch_bounds__(128)
void sim_topk_kernel(const float* __restrict__ A2,    // [B_ROWS, DIM] projected queries
                     const float* __restrict__ bank,  // [M_ROWS, DIM]
                     float* __restrict__ pvals,       // [B_ROWS, NSPLIT, 32]
                     int*   __restrict__ pidx)        // [B_ROWS, NSPLIT, 32]
{
    extern __shared__ char smem[];
    float* sA    = (float*)smem;                                   // TM*DIM        (256 KB)
    float* sB0   = sA + TM * DIM;                                  // NT*KCHUNK     ( 16 KB)
    float* sB1   = sB0 + NT * KCHUNK;                              // NT*KCHUNK     ( 16 KB)
    float* sTile = sB1 + NT * KCHUNK;                              // 4 waves * 256 (  4 KB)
    float* sTKV  = sTile + 4 * 256;                                // 64 rows * 32  (  8 KB)
    int*   sTKI  = (int*)(sTKV + TM * TOPK);                       // 64 rows * 32  (  8 KB)

    const int tid    = threadIdx.x;
    const int wave   = tid >> 5;
    const int lane   = tid & 31;
    const int l15    = lane & 15;
    const int khalf  = (lane >> 4) * 2;
    const int rowblk = blockIdx.x;          // which 64-row query block
    const int split  = blockIdx.y;          // which column stripe
    const int colbeg = split * STRIPE;

    // Async-load resident A block: TM*DIM floats = 16384 float4, 128 threads
    {
        const float* src = A2 + (size_t)rowblk * TM * DIM;
        for (int i = tid; i < TM * DIM / 4; i += 128)
            async_b128(src + (size_t)i * 4, sA + (size_t)i * 4);
    }
    // Init top-k lists while A loads are in flight
    for (int i = tid; i < TM * TOPK; i += 128) { sTKV[i] = NEG_INF; sTKI[i] = 0; }

    // Prologue: issue first bank chunk (q = 0 -> ct = 0, kc = 0) into sB0
    issue_bank_chunk(bank + (size_t)colbeg * DIM, sB0, tid);
    wait_async0();
    __syncthreads();

    float minV    = NEG_INF;  // threshold (32nd best) for this lane's row
    int   minSlot = 0;
    const float* myArow = sA + (size_t)(wave * 16 + l15) * DIM;

    const int NKC    = DIM / KCHUNK;                 // 4 chunks per column tile
    const int TOTALQ = (STRIPE / NT) * NKC;          // linearized chunk count
    int q = 0;

    for (int ct = 0; ct < STRIPE / NT; ++ct) {
        const int colbase = colbeg + ct * NT;
        v8f acc = {};
        for (int kcc = 0; kcc < NKC; ++kcc, ++q) {
            float* cur = (q & 1) ? sB1 : sB0;
            float* nxt = (q & 1) ? sB0 : sB1;
            if (q) {                       // q==0 already waited in prologue
                wait_async0();
                __syncthreads();
            }
            if (q + 1 < TOTALQ) {          // prefetch next chunk into other buffer
                const int nq  = q + 1;
                const int nct = nq >> 2;
                const int nkc = (nq & 3) * KCHUNK;
                issue_bank_chunk(bank + (size_t)(colbeg + nct * NT) * DIM + nkc, nxt, tid);
            }
            const int kc = kcc * KCHUNK;
#pragma unroll 8
            for (int k4 = 0; k4 < KCHUNK; k4 += 4) {
                const int kk = k4 + khalf;
                v2f a = *(const v2f*)(myArow + kc + kk);
                v2f b = *(const v2f*)(cur + (size_t)l15 * KCHUNK + kk);
                acc = __builtin_amdgcn_wmma_f32_16x16x4_f32(
                    false, a, false, b, (short)0, acc, false, false);
            }
        }
        // Spill 16x16 tile (per wave) to LDS for row-major scanning
        {
            const int mb = (lane >> 4) * 8;
#pragma unroll
            for (int j = 0; j < 8; ++j)
                sTile[wave * 256 + (mb + j) * 16 + l15] = acc[j];
        }
        // Top-k update: lanes 0..15 each own one query row of this wave
        if (lane < 16) {
            const float* rowv = sTile + wave * 256 + lane * 16;
            const int    base = (wave * 16 + lane) * TOPK;
#pragma unroll 4
            for (int c = 0; c < 16; ++c) {
                const float v = rowv[c];
                if (v > minV) {
                    sTKV[base + minSlot] = v;
                    sTKI[base + minSlot] = colbase + c;
                    // rescan to find new min
                    float mv = sTKV[base]; int ms = 0;
                    for (int t = 1; t < TOPK; ++t) {
                        const float tv = sTKV[base + t];
                        if (tv < mv) { mv = tv; ms = t; }
                    }
                    minV = mv; minSlot = ms;
                }
            }
        }
    }
    __syncthreads();

    // Emit partial lists: row gr gets NSPLIT partial lists of 32
    if (lane < 16) {
        const int gr   = rowblk * TM + wave * 16 + lane;
        const int base = (wave * 16 + lane) * TOPK;
        float* ov = pvals + ((size_t)gr * NSPLIT + split) * TOPK;
        int*   oi = pidx  + ((size_t)gr * NSPLIT + split) * TOPK;
        for (int t = 0; t < TOPK; ++t) { ov[t] = sTKV[base + t]; oi[t] = sTKI[base + t]; }
    }
}

// ---------------------------------------------------------------------------
// Merge NSPLIT partial lists per row into final descending top-32 indices.
// grid = B_ROWS/32, block = 32 (one thread per row)
// ---------------------------------------------------------------------------
__global__ __launch_bounds__(32)
void topk_merge_kernel(const float* __restrict__ pvals, const int* __restrict__ pidx,
                       int* __restrict__ fidx)
{
    __shared__ float cv[32][NSPLIT * TOPK];   // 16 KB
    __shared__ int   ci[32][NSPLIT * TOPK];   // 16 KB
    const int t   = threadIdx.x;
    const int row = blockIdx.x * 32 + t;
    for (int i = 0; i < NSPLIT * TOPK; ++i) {
        cv[t][i] = pvals[(size_t)row * NSPLIT * TOPK + i];
        ci[t][i] = pidx [(size_t)row * NSPLIT * TOPK + i];
    }
    for (int kk = 0; kk < TOPK; ++kk) {
        float best = NEG_INF; int bestI = 0x7fffffff; int bestSlot = 0;
        for (int i = 0; i < NSPLIT * TOPK; ++i) {
            const float v = cv[t][i];
            if (v > best || (v == best && ci[t][i] < bestI)) {
                best = v; bestI = ci[t][i]; bestSlot = i;
            }
        }
        fidx[(size_t)row * TOPK + kk] = bestI;
        cv[t][bestSlot] = NEG_INF; ci[t][bestSlot] = 0x7fffffff;
    }
}

// Zero the usage output region.
__global__ void zero_usage_kernel(float* __restrict__ usage)
{
    const int i = blockIdx.x * blockDim.x + threadIdx.x;
    if (i < M_ROWS) usage[i] = 0.0f;
}

// Gather retrieved rows + scatter-add usage.
// grid = B_ROWS*TOPK, block = 256 (float4 per thread)
__global__ __launch_bounds__(256)
void gather_scatter_kernel(const float* __restrict__ bank, const int* __restrict__ fidx,
                           float* __restrict__ out_ret, float* __restrict__ usage)
{
    const int row = blockIdx.x;             // 0 .. B_ROWS*TOPK-1
    const int idx = fidx[row];
    const float4* src = (const float4*)(bank + (size_t)idx * DIM);
    float4*       dst = (float4*)(out_ret + (size_t)row * DIM);
    dst[threadIdx.x] = src[threadIdx.x];
    if (threadIdx.x == 0) atomicAdd(usage + idx, 0.1f);
}

// ---------------------------------------------------------------------------
extern "C" void kernel_launch(void* const* d_in, const int* in_sizes, int n_in,
                              void* d_out, int out_size, void* d_ws, size_t ws_size,
                              hipStream_t stream)
{
    const float* query = (const float*)d_in[0];   // [B, D]
    const float* Wk    = (const float*)d_in[1];   // [D, D] (out,in)
    const float* bk    = (const float*)d_in[2];   // [D]
    const float* bank  = (const float*)d_in[3];   // [M, D]
    (void)d_in[4]; (void)in_sizes; (void)n_in; (void)out_size;

    // Workspace carve-up
    char*  ws    = (char*)d_ws;
    float* qk    = (float*)ws;                                       // 16 MB
    float* A2    = (float*)(ws + (size_t)B_ROWS * DIM * 4);          // 16 MB
    float* pvals = (float*)(ws + (size_t)B_ROWS * DIM * 8);          //  2 MB
    int*   pidx  = (int*)  (ws + (size_t)B_ROWS * DIM * 8
                               + (size_t)B_ROWS * NSPLIT * TOPK * 4);
    int*   fidx  = (int*)  (ws + (size_t)B_ROWS * DIM * 8
                               + (size_t)B_ROWS * NSPLIT * TOPK * 8);
    (void)ws_size;

    float* out_ret   = (float*)d_out;                                // [B, 32, D]
    float* out_usage = out_ret + (size_t)B_ROWS * TOPK * DIM;        // [M]

    // 1) qk = query @ Wk^T + bk
    proj_gemm_kernel<true, true><<<dim3(B_ROWS / 16, DIM / 64), 128, 0, stream>>>(
        query, Wk, bk, qk);
    // 2) A2 = qk @ Wk   (memory-side bias is a per-row constant -> drop)
    proj_gemm_kernel<false, false><<<dim3(B_ROWS / 16, DIM / 64), 128, 0, stream>>>(
        qk, Wk, nullptr, A2);
    // 3) sim = A2 @ bank^T fused with streaming per-row top-32
    {
        const size_t lds = ((size_t)TM * DIM + 2 * NT * KCHUNK + 4 * 256 + TM * TOPK) * 4
                         + (size_t)TM * TOPK * 4;
        sim_topk_kernel<<<dim3(B_ROWS / TM, NSPLIT), 128, lds, stream>>>(
            A2, bank, pvals, pidx);
    }
    // 4) merge partial lists into sorted-descending final indices
    topk_merge_kernel<<<B_ROWS / 32, 32, 0, stream>>>(pvals, pidx, fidx);
    // 5) usage = 0, then gather + scatter-add
    zero_usage_kernel<<<M_ROWS / 256, 256, 0, stream>>>(out_usage);
    gather_scatter_kernel<<<B_ROWS * TOPK, 256, 0, stream>>>(bank, fidx, out_ret, out_usage);
}